// CRF_51135880626938
// MI455X (gfx1250) — compile-verified
//
#include <hip/hip_runtime.h>
#include <stdint.h>

#define START_TAG 8
#define STOP_TAG  9
#define NEG_INF   -10000.0f
#define TT        10
#define CH        16
#define FWD_WAVES 4

typedef __attribute__((ext_vector_type(4))) unsigned int v4u;
typedef __attribute__((ext_vector_type(8))) int   v8i;
typedef __attribute__((ext_vector_type(4))) int   v4i;
typedef __attribute__((ext_vector_type(4))) float f4;
typedef __attribute__((ext_vector_type(2))) float f2;

// ---- Tensor Data Mover: 2D tile (tile_x elems per row, tile_y rows) -> LDS ----
__device__ __forceinline__ void tdm_load_2d(uint64_t gaddr, unsigned lds_off,
                                            unsigned tile_x, unsigned tile_y,
                                            unsigned row_stride_elems) {
  v4u g0;
  g0.x = 1u;                                        // count=1 valid descriptor
  g0.y = lds_off;                                   // LDS byte address
  g0.z = (unsigned)(gaddr & 0xFFFFFFFFu);           // global addr [31:0]
  g0.w = (unsigned)((gaddr >> 32) & 0x01FFFFFFu)    // global addr [56:32]
         | 0x80000000u;                             // type=2 ("image")
  v8i g1;
  g1[0] = (int)(2u << 16);                          // data_size=2 (4 bytes), wg_mask=0
  g1[1] = 0;                                        // tensor_dim0 lo16 (dim0 = 2^30)
  g1[2] = (int)0x00004000u;                         // tensor_dim0 hi16 ; tensor_dim1 lo16=0
  g1[3] = (int)(0x00004000u | (tile_x << 16));      // tensor_dim1 hi16 (2^30) | tile_dim0
  g1[4] = (int)(tile_y & 0xFFFFu);                  // tile_dim1 | tile_dim2=0
  g1[5] = (int)row_stride_elems;                    // tensor_dim0_stride lo32
  g1[6] = 0;
  g1[7] = 0;
  v4i gz = {0, 0, 0, 0};
#if __has_include(<hip/amd_detail/amd_gfx1250_TDM.h>)
  v8i gz8 = {0, 0, 0, 0, 0, 0, 0, 0};
  __builtin_amdgcn_tensor_load_to_lds(g0, g1, gz, gz, gz8, 0);
#else
  __builtin_amdgcn_tensor_load_to_lds(g0, g1, gz, gz, 0);
#endif
}

// ---------------- Forward Viterbi: 10 lanes per batch, 3 batches/wave --------
__global__ void __launch_bounds__(FWD_WAVES * 32)
crf_forward(const float* __restrict__ feats, const float* __restrict__ trans,
            const int* __restrict__ lengths, float* __restrict__ out_score,
            unsigned char* __restrict__ bptr, int* __restrict__ bestlast,
            int S, int B) {
  __shared__ __align__(16) float featbuf[FWD_WAVES][2][CH * 32];
  __shared__ __align__(16) float fvsh[FWD_WAVES][4 * 12];

  const int lane  = threadIdx.x & 31;
  const int w     = __builtin_amdgcn_readfirstlane((int)(threadIdx.x >> 5));
  const int waveg = blockIdx.x * FWD_WAVES + w;
  const int b0    = waveg * 3;
  if (b0 >= B) return;
  const int nb = (B - b0 < 3) ? (B - b0) : 3;
  const int g  = lane / TT;
  const int i  = lane - g * TT;
  const bool active = (lane < 30) && (g < nb);

  float tr[TT];
#pragma unroll
  for (int j = 0; j < TT; ++j) tr[j] = trans[i * TT + j];
  const float tstop = trans[STOP_TAG * TT + i];
  const int Lg = active ? lengths[b0 + g] : 0;

  float* fvp = &fvsh[w][g * 12];
  fvp[i] = (i == START_TAG) ? 0.0f : NEG_INF;   // fv_all[0]
  __builtin_amdgcn_wave_barrier();

  const int rowstride = B * TT;
  const int NCH = S / CH;
  const unsigned tilex = (unsigned)(nb * TT);
  const uint64_t gbase = (uint64_t)(uintptr_t)(feats + (size_t)b0 * TT);
  const uint64_t chunk_bytes = (uint64_t)CH * (uint64_t)rowstride * 4u;

  tdm_load_2d(gbase,               (unsigned)(uintptr_t)&featbuf[w][0][0], tilex, CH, (unsigned)rowstride);
  tdm_load_2d(gbase + chunk_bytes, (unsigned)(uintptr_t)&featbuf[w][1][0], tilex, CH, (unsigned)rowstride);

  float lfv = NEG_INF;
  const int fstride = nb * TT;

  for (int c = 0; c < NCH; ++c) {
    if (c == NCH - 1) __builtin_amdgcn_s_wait_tensorcnt(0);
    else              __builtin_amdgcn_s_wait_tensorcnt(1);
    asm volatile("" ::: "memory");
    const float* fb = &featbuf[w][c & 1][0];
#pragma unroll 4
    for (int t = 0; t < CH; ++t) {
      const int s = c * CH + t;
      f4 a  = *(const f4*)(fvp);
      f4 bq = *(const f4*)(fvp + 4);
      f2 cc = *(const f2*)(fvp + 8);
      const float feat = fb[t * fstride + lane];
      float m = a.x + tr[0]; int bi = 0; float v;
      v = a.y  + tr[1]; if (v > m) { m = v; bi = 1; }
      v = a.z  + tr[2]; if (v > m) { m = v; bi = 2; }
      v = a.w  + tr[3]; if (v > m) { m = v; bi = 3; }
      v = bq.x + tr[4]; if (v > m) { m = v; bi = 4; }
      v = bq.y + tr[5]; if (v > m) { m = v; bi = 5; }
      v = bq.z + tr[6]; if (v > m) { m = v; bi = 6; }
      v = bq.w + tr[7]; if (v > m) { m = v; bi = 7; }
      v = cc.x + tr[8]; if (v > m) { m = v; bi = 8; }
      v = cc.y + tr[9]; if (v > m) { m = v; bi = 9; }
      const float fn = m + feat;
      fvp[i] = fn;                                  // fv_all[s+1]
      lfv = (s + 1 == Lg) ? fn : lfv;               // capture fv at length
      if (active) bptr[((size_t)(b0 + g) * (size_t)S + (size_t)s) * 12 + i] = (unsigned char)bi;
      __builtin_amdgcn_wave_barrier();
    }
    if (c + 2 < NCH) {
      asm volatile("s_wait_dscnt 0" ::: "memory");  // LDS reads of this buffer drained
      tdm_load_2d(gbase + (uint64_t)(c + 2) * chunk_bytes,
                  (unsigned)(uintptr_t)&featbuf[w][c & 1][0], tilex, CH, (unsigned)rowstride);
    }
  }

  // terminal: score = max_i(last_fv[i] + trans[STOP][i]), best_last = argmax
  fvp[i] = lfv + tstop;
  __builtin_amdgcn_wave_barrier();
  if (active && i == 0) {
    float bm = fvp[0]; int bl = 0;
#pragma unroll
    for (int j = 1; j < TT; ++j) { float vv = fvp[j]; if (vv > bm) { bm = vv; bl = j; } }
    out_score[b0 + g] = bm;
    bestlast[b0 + g]  = bl;
  }
}

// ---------------- Backtrack: one wave per batch, LDS-staged pointer maps -----
__global__ void __launch_bounds__(256)
crf_backtrack(const unsigned char* __restrict__ bptr, const int* __restrict__ lengths,
              const int* __restrict__ bestlast, float* __restrict__ out_paths,
              int S, int B) {
  __shared__ __align__(4) unsigned char mbuf[8][32 * 12];
  const int lane = threadIdx.x & 31;
  const int w    = __builtin_amdgcn_readfirstlane((int)(threadIdx.x >> 5));
  const int b    = blockIdx.x * 8 + w;
  if (b >= B) return;
  const int L  = lengths[b];
  const int bl = bestlast[b];
  unsigned char* mb = &mbuf[w][0];
  const unsigned char* mrow = bptr + (size_t)b * (size_t)S * 12;

  int cur = bl;
  for (int S0 = S - 32; S0 >= 0; S0 -= 32) {
    int sidx = S0 + 1 + lane;                 // need map index s+1
    if (sidx > S - 1) sidx = S - 1;           // clamp (clamped slot never used)
    const unsigned* mp = (const unsigned*)(mrow + (size_t)sidx * 12);
    unsigned d0 = mp[0], d1 = mp[1], d2 = mp[2];
    unsigned* q = (unsigned*)(mb + lane * 12);
    q[0] = d0; q[1] = d1; q[2] = d2;
    __builtin_amdgcn_wave_barrier();          // LDS in-order within wave
    int myp = 0;
#pragma unroll
    for (int k = 31; k >= 0; --k) {
      const int s = S0 + k;
      const int nxt = (int)mb[k * 12 + cur];  // bptrs[s+1][b][cur]
      cur = (s == L - 1) ? bl : ((s < L - 1) ? nxt : cur);
      const int p = (s < L) ? cur : 0;
      if (k == lane) myp = p;
    }
    out_paths[(size_t)b * (size_t)S + (size_t)(S0 + lane)] = (float)myp;
    __builtin_amdgcn_wave_barrier();
  }
}

extern "C" void kernel_launch(void* const* d_in, const int* in_sizes, int n_in,
                              void* d_out, int out_size, void* d_ws, size_t ws_size,
                              hipStream_t stream) {
  const float* feats   = (const float*)d_in[0];
  const float* trans   = (const float*)d_in[1];
  const int*   lengths = (const int*)d_in[2];
  const int B = in_sizes[2];
  const int S = in_sizes[0] / (B * TT);
  float* out = (float*)d_out;

  int* bestlast = (int*)d_ws;                              // B ints
  unsigned char* bptr = (unsigned char*)d_ws + 16384;      // [b][s][12] bytes

  const int wavesNeeded = (B + 2) / 3;
  const int fblocks = (wavesNeeded + FWD_WAVES - 1) / FWD_WAVES;
  crf_forward<<<fblocks, FWD_WAVES * 32, 0, stream>>>(feats, trans, lengths,
                                                      out, bptr, bestlast, S, B);

  const int bblocks = (B + 7) / 8;
  crf_backtrack<<<bblocks, 256, 0, stream>>>(bptr, lengths, bestlast, out + B, S, B);
}